// GraphAttentionLayer_75368086110898
// MI455X (gfx1250) — compile-verified
//
#include <hip/hip_runtime.h>
#include <hip/hip_bf16.h>

// Fixed problem sizes from the reference harness.
constexpr int S = 4096;   // nodes / seq len
constexpr int F = 256;    // features
constexpr int B = 4;      // batch
constexpr int E = 4096;   // edges (== S, column index of mask == edge id)
constexpr int G = S / 16; // groups of 16 destination rows

typedef __attribute__((ext_vector_type(2))) float v2f;
typedef __attribute__((ext_vector_type(8))) float v8f;

// Workspace layout (ints): counts[S] | cursor[S] | offsets[S+1] | csr[2E+4]

// ---------------- CSR build ----------------

__global__ void k_zero(int* __restrict__ p) {          // zero counts + cursor
  int i = blockIdx.x * blockDim.x + threadIdx.x;
  if (i < 2 * S) p[i] = 0;
}

__global__ void k_count(const int* __restrict__ esrc, const int* __restrict__ edst,
                        int* __restrict__ counts) {
  int e = blockIdx.x * blockDim.x + threadIdx.x;
  if (e < E) {
    int s0 = esrc[e], d0 = edst[e];
    atomicAdd(&counts[s0], 1);
    if (d0 != s0) atomicAdd(&counts[d0], 1);   // set-union: count each column once
  }
}

__global__ __launch_bounds__(256) void k_scan(const int* __restrict__ counts,
                                              int* __restrict__ offsets,
                                              int* __restrict__ csr) {
  __shared__ int part[256];
  const int t = threadIdx.x;
  const int base = t * 16;
  int local[16];
  int s = 0;
#pragma unroll
  for (int q = 0; q < 16; ++q) { local[q] = s; s += counts[base + q]; }
  part[t] = s;
  __syncthreads();
  for (int off = 1; off < 256; off <<= 1) {
    int v = part[t];
    int add = (t >= off) ? part[t - off] : 0;
    __syncthreads();
    part[t] = v + add;
    __syncthreads();
  }
  int pre = (t == 0) ? 0 : part[t - 1];
#pragma unroll
  for (int q = 0; q < 16; ++q) offsets[base + q] = pre + local[q];
  if (t == 255) {
    int total = part[255];
    offsets[S] = total;
    // Sentinel pad: owner field = 4096 never matches a real row -> weight 0.
#pragma unroll
    for (int j = 0; j < 4; ++j) csr[total + j] = (S << 12);
  }
}

// O(E) scatter with atomic cursors. Intra-bucket order is nondeterministic here
// but k_sort makes the final CSR bit-deterministic (sorted by unique edge id).
__global__ void k_fill(const int* __restrict__ esrc, const int* __restrict__ edst,
                       const int* __restrict__ offsets, int* __restrict__ cursor,
                       int* __restrict__ csr) {
  int e = blockIdx.x * blockDim.x + threadIdx.x;
  if (e < E) {
    int s0 = esrc[e], d0 = edst[e];
    int ps = atomicAdd(&cursor[s0], 1);
    csr[offsets[s0] + ps] = e | (s0 << 12);
    if (d0 != s0) {
      int pd = atomicAdd(&cursor[d0], 1);
      csr[offsets[d0] + pd] = e | (d0 << 12);
    }
  }
}

// Per-bucket insertion sort by edge id (avg bucket size ~2). Restores the
// deterministic edge-id accumulation order for the fp sums downstream.
__global__ void k_sort(const int* __restrict__ offsets, int* __restrict__ csr) {
  int d = blockIdx.x * blockDim.x + threadIdx.x;
  if (d < S) {
    int o = offsets[d];
    int n = offsets[d + 1] - o;
    for (int i = 1; i < n; ++i) {
      int key = csr[o + i];
      int j = i - 1;
      while (j >= 0 && (csr[o + j] & (S - 1)) > (key & (S - 1))) {
        csr[o + j + 1] = csr[o + j];
        --j;
      }
      csr[o + j + 1] = key;
    }
  }
}

// ---------------- WMMA aggregation ----------------
// One wave per (batch b, group of 16 destination rows).
// out_tile(16 x 256) = A(16 x U) * X(U x 256), streamed in K-chunks of 4 slots
// through LDS, accumulated with V_WMMA_F32_16X16X4_F32 into 16 resident C tiles.
__global__ __launch_bounds__(32) void k_aggregate(const float* __restrict__ x,
                                                  const int* __restrict__ counts,
                                                  const int* __restrict__ offsets,
                                                  const int* __restrict__ csr,
                                                  float* __restrict__ out) {
  __shared__ __align__(16) float sx[4 * F];   // 4 staged source rows (4 KB)
  const int lane = threadIdx.x;
  const int g = blockIdx.x % G;
  const int b = blockIdx.x / G;
  const int i0 = g * 16;
  const int M = lane & 15;      // A row / C column index held by this lane
  const int half = lane >> 4;   // lanes 16-31 hold K=2,3 (A) / rows 2,3 (B) / M+8 (C)
  const int myRow = i0 + M;

  const int nb0 = offsets[i0];
  const int U = offsets[i0 + 16] - nb0;   // total slots for this group
  const int nchunk = (U + 3) >> 2;

  const int cnt = counts[myRow];
  const float invc = 1.0f / (float)(cnt > 0 ? cnt : 1);   // softmax == uniform over mask

  __builtin_prefetch(csr + nb0, 0, 1);    // gfx1250 global_prefetch_b8

  v8f acc[16] = {};                        // 16 feature tiles resident (128 VGPRs)
  const float* xb = x + (size_t)b * S * F;

  for (int c = 0; c < nchunk; ++c) {
    // Branchless slot fetch: CSR is sentinel-padded; overshoot entries belong to
    // a later bucket whose owner is outside this group -> weight 0, benign row.
    const int base = nb0 + 4 * c;
    const int k0 = csr[base + 0];
    const int k1 = csr[base + 1];
    const int k2 = csr[base + 2];
    const int k3 = csr[base + 3];
    const int s0 = k0 & (S - 1), s1 = k1 & (S - 1), s2 = k2 & (S - 1), s3 = k3 & (S - 1);
    const float w0 = ((k0 >> 12) == myRow) ? invc : 0.0f;
    const float w1 = ((k1 >> 12) == myRow) ? invc : 0.0f;
    const float w2 = ((k2 >> 12) == myRow) ? invc : 0.0f;
    const float w3 = ((k3 >> 12) == myRow) ? invc : 0.0f;

    // A-matrix (16x4 f32): VGPR0 holds K = 2*half, VGPR1 holds K = 2*half+1
    v2f a;
    a.x = half ? w2 : w0;
    a.y = half ? w3 : w1;

    __syncthreads();   // previous chunk's B reads done before restaging
#pragma unroll
    for (int t = 0; t < 8; ++t) {          // stage 4 rows x 256 f32 as float4
      const int row = t >> 1;              // uniform per unrolled step
      const int c4 = lane + ((t & 1) << 5);
      const int sr = (row == 0) ? s0 : (row == 1) ? s1 : (row == 2) ? s2 : s3;
      const float4 v = ((const float4*)(xb + (size_t)sr * F))[c4];
      ((float4*)sx)[row * 64 + c4] = v;
    }
    __syncthreads();

#pragma unroll
    for (int nt = 0; nt < 16; ++nt) {
      // B-matrix (4x16 f32): VGPR0 = row 2*half, VGPR1 = row 2*half+1, col = M
      v2f bb;
      bb.x = sx[(2 * half) * F + nt * 16 + M];
      bb.y = sx[(2 * half + 1) * F + nt * 16 + M];
      acc[nt] = __builtin_amdgcn_wmma_f32_16x16x4_f32(
          false, a, false, bb, (short)0, acc[nt], false, false);
    }
  }

  // C/D layout: VGPR r -> M = r + 8*half, N = lane&15
#pragma unroll
  for (int nt = 0; nt < 16; ++nt) {
#pragma unroll
    for (int r = 0; r < 8; ++r) {
      const int Mr = r + 8 * half;
      out[(size_t)(b * S + i0 + Mr) * F + nt * 16 + M] = acc[nt][r];
    }
  }
}

extern "C" void kernel_launch(void* const* d_in, const int* in_sizes, int n_in,
                              void* d_out, int out_size, void* d_ws, size_t ws_size,
                              hipStream_t stream) {
  (void)in_sizes; (void)n_in; (void)out_size; (void)ws_size;
  const float* x = (const float*)d_in[0];
  // d_in[1] = W: provably unused — softmax of a row-constant score is uniform
  // over the adjacency mask, so W cancels from the output.
  const int* esrc = (const int*)d_in[2];
  const int* edst = (const int*)d_in[3];
  float* out = (float*)d_out;

  int* counts = (int*)d_ws;            // S
  int* cursor = counts + S;            // S
  int* offsets = cursor + S;           // S+1
  int* csr = offsets + S + 1;          // 2E + 4 (sentinel pad)  (~80 KB total)

  k_zero<<<(2 * S) / 256, 256, 0, stream>>>(counts);
  k_count<<<E / 256, 256, 0, stream>>>(esrc, edst, counts);
  k_scan<<<1, 256, 0, stream>>>(counts, offsets, csr);
  k_fill<<<E / 256, 256, 0, stream>>>(esrc, edst, offsets, cursor, csr);
  k_sort<<<S / 256, 256, 0, stream>>>(offsets, csr);
  k_aggregate<<<B * G, 32, 0, stream>>>(x, counts, offsets, csr, out);
}